// Attention_17454747091547
// MI455X (gfx1250) — compile-verified
//
#include <hip/hip_runtime.h>

typedef float v2f __attribute__((ext_vector_type(2)));
typedef float v8f __attribute__((ext_vector_type(8)));

#define WAVES_PER_BLOCK 8
#define VROW 18   // padded LDS row stride (dwords): conflict-free, 8B-aligned pairs

// seg(s) packed as nibbles: s=0->0, 1..4->1, 5..8->2, 9->3, 10..15->4
#define SEG_PACKED 0x4444443222211110ull

// lane L <-> lane L^16 swap as a pure VALU op (v_permlanex16_b32, identity sels)
__device__ __forceinline__ float xor16(float v) {
  int i = __float_as_int(v);
  i = __builtin_amdgcn_permlanex16(i, i, 0x76543210, 0xfedcba98, false, false);
  return __int_as_float(i);
}

// raw v_exp_f32 (no libm range guards)
__device__ __forceinline__ float exp2_native(float v) {
  return __builtin_amdgcn_exp2f(v);
}

__global__ __launch_bounds__(256) void seg_attn_kernel(
    const float* __restrict__ x,
    const float* __restrict__ Wq, const float* __restrict__ bq,
    const float* __restrict__ Wk, const float* __restrict__ bk,
    const float* __restrict__ Wv, const float* __restrict__ bv,
    const float* __restrict__ Wo, const float* __restrict__ bo,
    float* __restrict__ out, int B)
{
  // per-wave transposed V' staging: 12 rows (2 heads x 6 outs) x VROW
  __shared__ __align__(16) float vs[WAVES_PER_BLOCK * 12 * VROW];

  const int tid  = threadIdx.x;
  const int lane = tid & 31;
  const int wv   = tid >> 5;
  const bool low = lane < 16;
  const int s    = lane & 15;               // row (query/key index) this lane owns
  const int g    = (int)((SEG_PACKED >> (4 * s)) & 0xF);
  const int o    = s;                       // output column this lane supplies in B/D
  const int oc   = (o > 5) ? 5 : o;         // clamped column for LDS reads
  const int kb   = low ? 0 : 2;             // B-operand k offset for this half
  const int hv   = low ? 0 : 1;             // head this lane's V' row belongs to
  float* vsw = vs + wv * 12 * VROW;

  const float scale = 0.57735026918962576451f; // 1/sqrt(3), folded into Q
  const float zm    = low ? 1.f : 0.f;         // zero-mask for dummy high-half rows

  // ---------------- per-lane weight registers (loaded once) ----------------
  // Q rows feed the WMMA B operand (Q^T), K rows feed the A operand.
  // j=0: head0 vgpr0 comp (low:0 / high:2); j=1: head0 vgpr1 (low:1 / high:zeroed)
  // j=2: head1 vgpr0 (low:3 / high:5);      j=3: head1 vgpr1 (low:4 / high:zeroed)
  int qcomp[4];
  qcomp[0] = low ? 0 : 2;
  qcomp[1] = low ? 1 : 2;
  qcomp[2] = low ? 3 : 5;
  qcomp[3] = low ? 4 : 5;

  float wqr[4][6], bqr[4], wkr[4][6], bkr[4];
#pragma unroll
  for (int j = 0; j < 4; ++j) {
    const float msk = (j == 1 || j == 3) ? zm : 1.f;  // dummy rows -> exact zeros
    const float* wqp = Wq + g * 36 + qcomp[j] * 6;
    const float* wkp = Wk + g * 36 + qcomp[j] * 6;
#pragma unroll
    for (int e = 0; e < 6; ++e) {
      wqr[j][e] = wqp[e] * scale * msk;
      wkr[j][e] = wkp[e] * msk;
    }
    bqr[j] = bq[g * 6 + qcomp[j]] * scale * msk;
    bkr[j] = bk[g * 6 + qcomp[j]] * msk;
  }

  // Fused V*Wo^T weights for this lane's head. Biases (bv via Wo, plus bo/2 per
  // head) fold into every V' row because softmax rows sum to 1.
  float wvo[6][6], bvo[6];
#pragma unroll
  for (int oo = 0; oo < 6; ++oo) {
    float bacc = 0.5f * bo[oo];
#pragma unroll
    for (int d = 0; d < 3; ++d)
      bacc += Wo[oo * 6 + 3 * hv + d] * bv[g * 6 + 3 * hv + d];
    bvo[oo] = bacc;
#pragma unroll
    for (int e = 0; e < 6; ++e) {
      float acc = 0.f;
#pragma unroll
      for (int d = 0; d < 3; ++d)
        acc += Wo[oo * 6 + 3 * hv + d] * Wv[g * 36 + (3 * hv + d) * 6 + e];
      wvo[oo][e] = acc;
    }
  }

  // loop-invariant LDS pointers (immediate offsets inside the loop)
  float* vst = vsw + hv * 6 * VROW + s;                 // V' store base
  const float* vb0 = vsw + (0 * 6 + oc) * VROW + kb;    // head0 B-operand base
  const float* vb1 = vsw + (1 * 6 + oc) * VROW + kb;    // head1 B-operand base

  const int gw = blockIdx.x * WAVES_PER_BLOCK + wv;
  const int nw = gridDim.x * WAVES_PER_BLOCK;

  const v8f vzero = {0.f, 0.f, 0.f, 0.f, 0.f, 0.f, 0.f, 0.f};
  const float L2E = 1.4426950408889634f;

  for (int b = gw; b < B; b += nw) {
    const float* xb = x + (size_t)b * 96;
    if (b + nw < B)
      __builtin_prefetch(x + (size_t)(b + nw) * 96, 0, 1);

    // own input row (24B, 8-byte aligned -> b64 pairs)
    float xr[6];
    const float* xrow = xb + s * 6;
#pragma unroll
    for (int e = 0; e < 3; ++e) {
      float2 t = *reinterpret_cast<const float2*>(xrow + 2 * e);
      xr[2 * e] = t.x; xr[2 * e + 1] = t.y;
    }

    // ---- projections (VALU, weights in registers; dummies already zeroed) ----
    float qv[4], kv[4];
#pragma unroll
    for (int j = 0; j < 4; ++j) {
      float aq = bqr[j], ak = bkr[j];
#pragma unroll
      for (int e = 0; e < 6; ++e) { aq += wqr[j][e] * xr[e]; ak += wkr[j][e] * xr[e]; }
      qv[j] = aq; kv[j] = ak;
    }

    // ---- fused V' row -> transposed LDS tile ----
#pragma unroll
    for (int oo = 0; oo < 6; ++oo) {
      float acc = bvo[oo];
#pragma unroll
      for (int e = 0; e < 6; ++e) acc += wvo[oo][e] * xr[e];
      vst[oo * VROW] = acc;
    }

    // ---- scores^T = K_h * Q_h^T per head : one WMMA f32 16x16x4 each ----
    v2f a, bb;
    a[0] = kv[0]; a[1] = kv[1]; bb[0] = qv[0]; bb[1] = qv[1];
    v8f st0 = __builtin_amdgcn_wmma_f32_16x16x4_f32(false, a, false, bb,
                                                    (short)0, vzero, false, false);
    a[0] = kv[2]; a[1] = kv[3]; bb[0] = qv[2]; bb[1] = qv[3];
    v8f st1 = __builtin_amdgcn_wmma_f32_16x16x4_f32(false, a, false, bb,
                                                    (short)0, vzero, false, false);

    // ---- softmax over k (M dim), both heads interleaved for TRANS latency ----
    // exp(t - m) computed as v_fma(t, log2e, -m*log2e) + raw v_exp_f32
    v8f att0, att1;
    {
      float m0 = st0[0], m1 = st1[0];
#pragma unroll
      for (int r = 1; r < 8; ++r) { m0 = fmaxf(m0, st0[r]); m1 = fmaxf(m1, st1[r]); }
      m0 = fmaxf(m0, xor16(m0));
      m1 = fmaxf(m1, xor16(m1));
      const float n0 = m0 * L2E, n1 = m1 * L2E;
      float s0 = 0.f, s1 = 0.f;
#pragma unroll
      for (int r = 0; r < 8; ++r) {
        att0[r] = exp2_native(fmaf(st0[r], L2E, -n0)); s0 += att0[r];
        att1[r] = exp2_native(fmaf(st1[r], L2E, -n1)); s1 += att1[r];
      }
      s0 += xor16(s0);
      s1 += xor16(s1);
      const float r0 = __frcp_rn(s0), r1 = __frcp_rn(s1);
#pragma unroll
      for (int r = 0; r < 8; ++r) { att0[r] *= r0; att1[r] *= r1; }
    }

    // ---- out tile = sum_h attn_h @ V'_h : 8 chained WMMAs into one D ----
    v8f d = vzero;
#pragma unroll
    for (int h = 0; h < 2; ++h) {
      const v8f at = h ? att1 : att0;
      const float* vbh = h ? vb1 : vb0;

      // cross-half copies (C-layout -> A-layout fixup), pure VALU permlanes
      float sxv[8];
#pragma unroll
      for (int r = 0; r < 8; ++r) sxv[r] = xor16(at[r]);

      float a0c[4], a1c[4];
      a0c[0] = low ? at[0]  : sxv[2];  a1c[0] = low ? at[1]  : sxv[3];  // k=0..3
      a0c[1] = low ? at[4]  : sxv[6];  a1c[1] = low ? at[5]  : sxv[7];  // k=4..7
      a0c[2] = low ? sxv[0] : at[2];   a1c[2] = low ? sxv[1] : at[3];   // k=8..11
      a0c[3] = low ? sxv[4] : at[6];   a1c[3] = low ? sxv[5] : at[7];   // k=12..15

#pragma unroll
      for (int c = 0; c < 4; ++c) {
        v2f av, bv2;
        av[0] = a0c[c]; av[1] = a1c[c];
        float2 bp = *reinterpret_cast<const float2*>(vbh + 4 * c);  // ds_load_b64
        bv2[0] = bp.x; bv2[1] = bp.y;
        d = __builtin_amdgcn_wmma_f32_16x16x4_f32(false, av, false, bv2,
                                                  (short)0, d, false, false);
      }
    }

    // ---- store D tile directly (biases already folded via V') ----
    if (o < 6) {
      float* ob = out + (size_t)b * 96 + (low ? 0 : 48) + o;
#pragma unroll
      for (int r = 0; r < 8; ++r) ob[r * 6] = d[r];
    }
  }
}

extern "C" void kernel_launch(void* const* d_in, const int* in_sizes, int n_in,
                              void* d_out, int out_size, void* d_ws, size_t ws_size,
                              hipStream_t stream) {
  (void)n_in; (void)out_size; (void)d_ws; (void)ws_size;
  const float* x  = (const float*)d_in[0];
  const float* Wq = (const float*)d_in[1];
  const float* bq = (const float*)d_in[2];
  const float* Wk = (const float*)d_in[3];
  const float* bk = (const float*)d_in[4];
  const float* Wv = (const float*)d_in[5];
  const float* bv = (const float*)d_in[6];
  const float* Wo = (const float*)d_in[7];
  const float* bo = (const float*)d_in[8];
  float* out = (float*)d_out;

  const int B = in_sizes[0] / 96;           // x is [B,16,6]
  int blocks = 2048;                        // 16384 waves -> ~16 batches/wave
  const int nw = blocks * WAVES_PER_BLOCK;
  if (nw > B) blocks = (B + WAVES_PER_BLOCK - 1) / WAVES_PER_BLOCK;
  if (blocks < 1) blocks = 1;

  seg_attn_kernel<<<blocks, 256, 0, stream>>>(x, Wq, bq, Wk, bk, Wv, bv, Wo, bo,
                                              out, B);
}